// ResonantFFNV2_75892072121014
// MI455X (gfx1250) — compile-verified
//
#include <hip/hip_runtime.h>
#include <hip/hip_bf16.h>

// Problem constants (from reference setup_inputs)
#define BB 2
#define SS 512
#define DD 512
#define NN 256
#define BS (BB * SS)          // 1024 rows
#define LUT_K 651.8986469044033f   // 4096 / (2*pi)
#define INV_LUT 0.000244140625f    // 1 / 4096

typedef float v2f __attribute__((ext_vector_type(2)));
typedef float v8f __attribute__((ext_vector_type(8)));

// ---------------------------------------------------------------------------
// Kernel 0: rw[n,d] = 1 / (1 + |W[n,d]|)
// ---------------------------------------------------------------------------
__global__ void __launch_bounds__(256) k_prep_rw(const float* __restrict__ W,
                                                 float* __restrict__ rw) {
    int i = blockIdx.x * 256 + threadIdx.x;
    if (i < NN * DD) {
        rw[i] = 1.0f / (1.0f + fabsf(W[i]));
    }
}

// ---------------------------------------------------------------------------
// Kernel 1: cos_sum/sin_sum[bs,n] = sum_d {cos,sin}(theta)*attn_{cos,sin}[n,d]
// theta = x[bs,d]*rw[n,d] + B_param[n,d]; LUT quantization reproduced exactly
// via v_sin/v_cos (hardware takes revolutions): sin(2*pi * idx/4096).
// Block: 256 threads = one n per thread; BSTILE (b,s) rows per block with x
// rows staged in LDS; tables stream from L2 (2 MB total, L2-resident).
// ---------------------------------------------------------------------------
#define BSTILE 4

__device__ __forceinline__ void res_accum(float xv, float r, float bp,
                                          float ac, float as_,
                                          float& accc, float& accs) {
    float theta = fmaf(xv, r, bp);
    // idx = rint(theta * 4096/(2pi));  frac = idx/4096;  sin(2pi*frac)
    float fr = __builtin_rintf(theta * LUT_K) * INV_LUT;
    accc = fmaf(__builtin_amdgcn_cosf(fr), ac, accc);
    accs = fmaf(__builtin_amdgcn_sinf(fr), as_, accs);
}

__global__ void __launch_bounds__(256) k_stage1(const float* __restrict__ x,
                                                const float* __restrict__ rw,
                                                const float* __restrict__ Bp,
                                                const float* __restrict__ acns,
                                                const float* __restrict__ asns,
                                                float* __restrict__ cs,
                                                float* __restrict__ ss) {
    __shared__ float xs[BSTILE][DD];
    const int bs0 = blockIdx.x * BSTILE;

    // stage x rows into LDS
    for (int i = threadIdx.x; i < BSTILE * DD; i += 256) {
        ((float*)xs)[i] = x[bs0 * DD + i];
    }
    __syncthreads();

    const int n = threadIdx.x;                 // one n per thread (NN == 256)
    const float4* rw4 = (const float4*)(rw   + n * DD);
    const float4* bp4 = (const float4*)(Bp   + n * DD);
    const float4* ac4 = (const float4*)(acns + n * DD);
    const float4* as4 = (const float4*)(asns + n * DD);

    float accc[BSTILE], accs[BSTILE];
#pragma unroll
    for (int t = 0; t < BSTILE; ++t) { accc[t] = 0.0f; accs[t] = 0.0f; }

    for (int d4 = 0; d4 < DD / 4; ++d4) {
        float4 r  = rw4[d4];
        float4 bp = bp4[d4];
        float4 c4 = ac4[d4];
        float4 s4 = as4[d4];
#pragma unroll
        for (int t = 0; t < BSTILE; ++t) {
            float4 xv = ((const float4*)xs[t])[d4];
            res_accum(xv.x, r.x, bp.x, c4.x, s4.x, accc[t], accs[t]);
            res_accum(xv.y, r.y, bp.y, c4.y, s4.y, accc[t], accs[t]);
            res_accum(xv.z, r.z, bp.z, c4.z, s4.z, accc[t], accs[t]);
            res_accum(xv.w, r.w, bp.w, c4.w, s4.w, accc[t], accs[t]);
        }
    }

#pragma unroll
    for (int t = 0; t < BSTILE; ++t) {
        cs[(bs0 + t) * NN + n] = accc[t];
        ss[(bs0 + t) * NN + n] = accs[t];
    }
}

// ---------------------------------------------------------------------------
// Kernel 2: out[bs,d] = silu( sum_n cs[bs,n]*pc[d,n] + ss[bs,n]*ps[d,n] )
// f32 GEMM via V_WMMA_F32_16X16X4_F32. One wave per 16x16 output tile,
// K = NN = 256 -> 64 k-steps x 2 WMMAs (cos + sin chained into one acc).
//
// Operand layouts (wave32):
//  A 16x4 f32 : lane L(0..15): v0=A[L][0], v1=A[L][1]; lane L+16: v0=A[L][2], v1=A[L][3]
//  B 4x16 f32 : lane L(0..15): v0=B[0][L], v1=B[1][L]; lane L+16: v0=B[2][L], v1=B[3][L]
//  C/D 16x16  : VGPR i: lanes 0-15 -> (M=i, N=lane); lanes 16-31 -> (M=i+8, N=lane-16)
// ---------------------------------------------------------------------------
__global__ void __launch_bounds__(256) k_stage2(const float* __restrict__ cs,
                                                const float* __restrict__ ss,
                                                const float* __restrict__ pc,
                                                const float* __restrict__ ps,
                                                float* __restrict__ out) {
    const int wave = threadIdx.x >> 5;                 // 0..7
    const int lane = threadIdx.x & 31;
    const int tile = blockIdx.x * 8 + wave;            // 0..2047
    const int tm = tile >> 5;                          // 64 M-tiles
    const int tn = tile & 31;                          // 32 N-tiles
    const int m0 = tm * 16;
    const int d0 = tn * 16;

    const int ml    = lane & 15;
    const int khalf = lane >> 4;                       // selects K-pair {0,1} vs {2,3}

    const float* csrow = cs + (m0 + ml) * NN;          // A rows (cos)
    const float* ssrow = ss + (m0 + ml) * NN;          // A rows (sin)
    const float* pcrow = pc + (d0 + ml) * NN;          // B cols come from proj rows
    const float* psrow = ps + (d0 + ml) * NN;

    v8f acc = {};
    for (int k = 0; k < NN; k += 4) {
        const int o = k + 2 * khalf;
        v2f a_c = *(const v2f*)(csrow + o);
        v2f b_c = *(const v2f*)(pcrow + o);
        acc = __builtin_amdgcn_wmma_f32_16x16x4_f32(
            false, a_c, false, b_c, (short)0, acc, false, false);
        v2f a_s = *(const v2f*)(ssrow + o);
        v2f b_s = *(const v2f*)(psrow + o);
        acc = __builtin_amdgcn_wmma_f32_16x16x4_f32(
            false, a_s, false, b_s, (short)0, acc, false, false);
    }

    // SiLU epilogue + store (D-matrix layout above)
#pragma unroll
    for (int i = 0; i < 8; ++i) {
        const int m  = m0 + i + 8 * khalf;
        const int dd = d0 + ml;
        float v = acc[i];
        float sig = 1.0f / (1.0f + __expf(-v));
        out[m * DD + dd] = v * sig;
    }
}

// ---------------------------------------------------------------------------
// Launcher
// ---------------------------------------------------------------------------
extern "C" void kernel_launch(void* const* d_in, const int* in_sizes, int n_in,
                              void* d_out, int out_size, void* d_ws, size_t ws_size,
                              hipStream_t stream) {
    const float* x   = (const float*)d_in[0];   // (B,S,D)
    const float* W   = (const float*)d_in[1];   // (N,D)
    const float* Bp  = (const float*)d_in[2];   // (N,D)
    const float* ac  = (const float*)d_in[3];   // (N,D)
    const float* as_ = (const float*)d_in[4];   // (N,D)
    const float* pc  = (const float*)d_in[5];   // (D,N)
    const float* ps  = (const float*)d_in[6];   // (D,N)
    float* out = (float*)d_out;                 // (B,S,D) f32

    float* rw = (float*)d_ws;                   // N*D
    float* cs = rw + NN * DD;                   // BS*N
    float* ss = cs + BS * NN;                   // BS*N

    // Kernel 0: reciprocal wavelength
    k_prep_rw<<<(NN * DD + 255) / 256, 256, 0, stream>>>(W, rw);

    // Kernel 1: resonant sums (BS/BSTILE blocks, one n per thread)
    k_stage1<<<BS / BSTILE, 256, 0, stream>>>(x, rw, Bp, ac, as_, cs, ss);

    // Kernel 2: dual f32 WMMA GEMM + SiLU (2048 tiles, 8 waves/block)
    k_stage2<<<(BS / 16) * (DD / 16) / 8, 256, 0, stream>>>(cs, ss, pc, ps, out);
}